// Decoder_13889924235715
// MI455X (gfx1250) — compile-verified
//
#include <hip/hip_runtime.h>
#include <math.h>

typedef __attribute__((ext_vector_type(2))) float v2f;
typedef __attribute__((ext_vector_type(8))) float v8f;

#define HDIM 1024
#define VDIM 50257
#define LDIM 4096

#if __has_builtin(__builtin_amdgcn_global_load_async_to_lds_b32)
#define USE_ASYNC_LDS 1
typedef __attribute__((address_space(1))) int as1i;   // global int
typedef __attribute__((address_space(3))) int as3i;   // LDS int
#endif

__device__ __forceinline__ void wait_async_zero() {
#if __has_builtin(__builtin_amdgcn_s_wait_asynccnt)
    __builtin_amdgcn_s_wait_asynccnt(0);
#else
    asm volatile("s_wait_asynccnt 0x0" ::: "memory");
#endif
}

// Build attn_in = [emb[x], h0] (2048) and cat[0:H] = emb[x]
__global__ void prep_kernel(const int* __restrict__ x,
                            const float* __restrict__ h,
                            const float* __restrict__ emb,
                            float* __restrict__ attn_in,
                            float* __restrict__ cat) {
    int i = blockIdx.x * blockDim.x + threadIdx.x;
    if (i >= 2 * HDIM) return;
    int xi = x[0];
    if (i < HDIM) {
        float e = emb[(size_t)xi * HDIM + i];
        attn_in[i] = e;
        cat[i] = e;
    } else {
        attn_in[i] = h[i - HDIM];
    }
}

// GEMV via V_WMMA_F32_16X16X4_F32: out[m] = dot(W[m,:], vec) + bias[m]
// One wave per 16-row tile. A tile (16x4 fp32) maps to one b64 load per lane
// (lanes 0-15: K={0,1}, lanes 16-31: K={2,3}); B columns broadcast the vector
// so every D column carries the same dot products (lane 0 -> rows 0..7,
// lane 16 -> rows 8..15).
__launch_bounds__(256)
__global__ void gemv_wmma_kernel(const float* __restrict__ W,
                                 const float* __restrict__ bias,
                                 const float* __restrict__ vec,
                                 float* __restrict__ out,
                                 int M, int K, int relu_act) {
    __shared__ float s_vec[2048];
    int tid = threadIdx.x;
#ifdef USE_ASYNC_LDS
    // ASYNCcnt-tracked direct global->LDS staging (no VGPR bounce)
    for (int i = tid; i < K; i += 256) {
        __builtin_amdgcn_global_load_async_to_lds_b32(
            (as1i*)(vec + i), (as3i*)(&s_vec[i]), 0, 0);
    }
    wait_async_zero();
    __syncthreads();
#else
    for (int i = tid; i < K; i += 256) s_vec[i] = vec[i];
    __syncthreads();
#endif

    int wave = tid >> 5;                 // 8 waves / block
    int lane = tid & 31;
    int tile = blockIdx.x * 8 + wave;
    int row0 = tile * 16;
    if (row0 >= M) return;               // wave-uniform exit (after barrier)

    int r    = lane & 15;                // A-matrix row within tile
    int half = lane >> 4;                // 0 -> K pair {0,1}, 1 -> {2,3}
    int row  = row0 + r;
    int rowc = row < M ? row : (M - 1);  // clamp; clamped results discarded
    const float* wrow = W + (size_t)rowc * K + 2 * half;

    v8f acc = {0.f, 0.f, 0.f, 0.f, 0.f, 0.f, 0.f, 0.f};
#pragma unroll 8
    for (int k = 0; k < K; k += 4) {
        // speculative prefetch ~2KB ahead in this row (safe past-end)
        __builtin_prefetch(wrow + k + 512, 0, 3);
        v2f a = *(const v2f*)(wrow + k);               // A[m, k+2h .. k+2h+1]
        v2f b = *(const v2f*)(&s_vec[k + 2 * half]);   // vec broadcast into B cols
        acc = __builtin_amdgcn_wmma_f32_16x16x4_f32(
            false, a, false, b, (short)0, acc, false, false);
    }

    if (r == 0) {                        // lanes 0 and 16 hold the 16 rows
        int base = row0 + half * 8;
#pragma unroll
        for (int m = 0; m < 8; ++m) {
            int rr = base + m;
            if (rr < M) {
                float v = acc[m] + bias[rr];
                if (relu_act) v = v > 0.f ? v : 0.f;
                out[rr] = v;
            }
        }
    }
}

// In-place log_softmax over 4096 values, one 1024-thread block.
__launch_bounds__(1024)
__global__ void logsoftmax_kernel(float* __restrict__ a) {
    __shared__ float sm[1024];
    int tid = threadIdx.x;
    float v0 = a[tid], v1 = a[tid + 1024], v2 = a[tid + 2048], v3 = a[tid + 3072];
    float m = fmaxf(fmaxf(v0, v1), fmaxf(v2, v3));
    sm[tid] = m; __syncthreads();
    for (int s = 512; s > 0; s >>= 1) {
        if (tid < s) sm[tid] = fmaxf(sm[tid], sm[tid + s]);
        __syncthreads();
    }
    float gmax = sm[0];
    __syncthreads();
    float e = expf(v0 - gmax) + expf(v1 - gmax) + expf(v2 - gmax) + expf(v3 - gmax);
    sm[tid] = e; __syncthreads();
    for (int s = 512; s > 0; s >>= 1) {
        if (tid < s) sm[tid] += sm[tid + s];
        __syncthreads();
    }
    float lse = gmax + logf(sm[0]);
    a[tid]        = v0 - lse;
    a[tid + 1024] = v1 - lse;
    a[tid + 2048] = v2 - lse;
    a[tid + 3072] = v3 - lse;
}

// partial[by][col] = sum over a 256-row slice of w[l] * enc[l][col]
__launch_bounds__(256)
__global__ void attn_apply_partial_kernel(const float* __restrict__ w,
                                          const float* __restrict__ enc,
                                          float* __restrict__ partial) {
    int col = blockIdx.x * 256 + threadIdx.x;   // gridDim.x = H/256 = 4
    int l0  = blockIdx.y * 256;                 // gridDim.y = 16 slices of L
    float acc = 0.f;
#pragma unroll 4
    for (int l = 0; l < 256; ++l) {
        __builtin_prefetch(enc + (size_t)(l0 + l + 8) * HDIM + col, 0, 3);
        acc = fmaf(w[l0 + l], enc[(size_t)(l0 + l) * HDIM + col], acc);
    }
    partial[blockIdx.y * HDIM + col] = acc;
}

__launch_bounds__(256)
__global__ void attn_reduce_kernel(const float* __restrict__ partial,
                                   float* __restrict__ cat_hi) {
    int col = blockIdx.x * 256 + threadIdx.x;
    float s = 0.f;
#pragma unroll
    for (int p = 0; p < 16; ++p) s += partial[p * HDIM + col];
    cat_hi[col] = s;
}

__launch_bounds__(256)
__global__ void gru_gate_kernel(const float* __restrict__ gi,
                                const float* __restrict__ gh,
                                const float* __restrict__ h,
                                float* __restrict__ h_new) {
    int i = blockIdx.x * 256 + threadIdx.x;
    if (i >= HDIM) return;
    float ir = gi[i], iz = gi[HDIM + i], in_ = gi[2 * HDIM + i];
    float hr = gh[i], hz = gh[HDIM + i], hn  = gh[2 * HDIM + i];
    float rr = 1.f / (1.f + expf(-(ir + hr)));
    float zz = 1.f / (1.f + expf(-(iz + hz)));
    float nn = tanhf(in_ + rr * hn);
    h_new[i] = (1.f - zz) * nn + zz * h[i];
}

extern "C" void kernel_launch(void* const* d_in, const int* in_sizes, int n_in,
                              void* d_out, int out_size, void* d_ws, size_t ws_size,
                              hipStream_t stream) {
    const int*   x      = (const int*)  d_in[0];
    const float* h      = (const float*)d_in[1];
    const float* enc    = (const float*)d_in[2];
    const float* emb    = (const float*)d_in[3];
    const float* attn_W = (const float*)d_in[4];
    const float* attn_b = (const float*)d_in[5];
    const float* ctx_W  = (const float*)d_in[6];
    const float* ctx_b  = (const float*)d_in[7];
    const float* W_ih   = (const float*)d_in[8];
    const float* W_hh   = (const float*)d_in[9];
    const float* b_ih   = (const float*)d_in[10];
    const float* b_hh   = (const float*)d_in[11];
    const float* out_W  = (const float*)d_in[12];
    const float* out_b  = (const float*)d_in[13];
    float* out = (float*)d_out;
    float* ws  = (float*)d_ws;

    // workspace layout (floats), total 32768 = 128 KB
    float* attn_in = ws;             // 2048
    float* attn_lg = ws + 2048;      // 4096 -> log-softmax weights in place
    float* partial = ws + 6144;      // 16*1024
    float* cat     = ws + 22528;     // 2048
    float* g       = ws + 24576;     // 1024
    float* gi      = ws + 25600;     // 3072
    float* gh      = ws + 28672;     // 3072
    float* h_new   = ws + 31744;     // 1024

    prep_kernel<<<8, 256, 0, stream>>>(x, h, emb, attn_in, cat);

    // attn logits: 4096 x 2048
    gemv_wmma_kernel<<<(LDIM / 16 + 7) / 8, 256, 0, stream>>>(
        attn_W, attn_b, attn_in, attn_lg, LDIM, 2 * HDIM, 0);

    logsoftmax_kernel<<<1, 1024, 0, stream>>>(attn_lg);

    attn_apply_partial_kernel<<<dim3(4, 16), 256, 0, stream>>>(attn_lg, enc, partial);
    attn_reduce_kernel<<<4, 256, 0, stream>>>(partial, cat + HDIM);

    // g = relu(ctx_W @ cat + ctx_b): 1024 x 2048
    gemv_wmma_kernel<<<(HDIM / 16 + 7) / 8, 256, 0, stream>>>(
        ctx_W, ctx_b, cat, g, HDIM, 2 * HDIM, 1);

    // gi = W_ih @ g + b_ih ; gh = W_hh @ h0 + b_hh : 3072 x 1024 each
    gemv_wmma_kernel<<<(3 * HDIM / 16 + 7) / 8, 256, 0, stream>>>(
        W_ih, b_ih, g, gi, 3 * HDIM, HDIM, 0);
    gemv_wmma_kernel<<<(3 * HDIM / 16 + 7) / 8, 256, 0, stream>>>(
        W_hh, b_hh, h, gh, 3 * HDIM, HDIM, 0);

    gru_gate_kernel<<<4, 256, 0, stream>>>(gi, gh, h, h_new);

    // logits = out_W @ h_new + out_b : 50257 x 1024  (206 MB stream, the hot one)
    int tiles = (VDIM + 15) / 16;
    gemv_wmma_kernel<<<(tiles + 7) / 8, 256, 0, stream>>>(
        out_W, out_b, h_new, out, VDIM, HDIM, 0);
}